// VectorQuantizer_85452669321611
// MI455X (gfx1250) — compile-verified
//
#include <hip/hip_runtime.h>
#include <hip/hip_bf16.h>

typedef __attribute__((ext_vector_type(2))) float v2f;
typedef __attribute__((ext_vector_type(8))) float v8f;
typedef __attribute__((ext_vector_type(4))) unsigned int v4u;
typedef __attribute__((ext_vector_type(8))) int v8i;
typedef __attribute__((ext_vector_type(4))) int v4i;

// Problem constants (from reference)
constexpr int NPOS   = 16384;          // 16*32*32 spatial vectors
constexpr int KCODE  = 8192;           // codebook size
constexpr int CDIM   = 256;            // embedding dim
constexpr int QELEMS = 16 * 256 * 32 * 32;   // 4194304
constexpr int IDX_OFF = QELEMS + 2;          // d_out: [quantized | cb_loss | commit_loss | indices]
constexpr int NTILE  = KCODE / 16;           // 512 code tiles
constexpr int LDSROW = 260;                  // padded row stride (floats): conflict-free, 16B-aligned

// ---------------- Kernel 1: ||e_k||^2 ----------------
__global__ void vq_enorm(const float* __restrict__ emb, float* __restrict__ enorm) {
    int lane = threadIdx.x & 31;
    int wave = threadIdx.x >> 5;
    int r = blockIdx.x * 8 + wave;            // 1024 blocks * 8 waves = 8192 rows
    const float* row = emb + (size_t)r * CDIM;
    float s = 0.f;
#pragma unroll
    for (int k = 0; k < 8; ++k) { float v = row[lane + 32 * k]; s += v * v; }
#pragma unroll
    for (int mm = 16; mm >= 1; mm >>= 1) s += __shfl_xor(s, mm, 32);
    if (lane == 0) enorm[r] = s;
}

// ---------------- TDM: DMA one 16x256 f32 emb tile into padded LDS rows ----------------
// D# per CDNA5 ISA. pad_interval=7 (every 256 DWORDs) +
// pad_amount=3 (4 DWORDs) => LDS row stride 260 floats (bank-conflict-free).
// This toolchain exposes the 6-arg builtin: (v4u, v8i, v4i, v4i, v8i, i32 cpol).
#if __has_builtin(__builtin_amdgcn_tensor_load_to_lds)
#define HAVE_TDM 1
__device__ __forceinline__ void tdm_load_tile(const float* gsrc, unsigned lds_off) {
    unsigned long long ga = (unsigned long long)(uintptr_t)gsrc;
    v4u g0;
    g0.x = 1u;                                            // count=1 (valid), no gather
    g0.y = lds_off;                                       // lds_addr (bytes)
    g0.z = (unsigned)(ga & 0xFFFFFFFFu);                  // global_addr[31:0]
    g0.w = (unsigned)((ga >> 32) & 0x01FFFFFFu)           // global_addr[56:32]
         | 0x80000000u;                                   // type=2 ("image")
    v8i g1;
    g1[0] = (int)((2u << 16)        // data_size = 4 bytes
                | (1u << 20)        // pad_enable
                | (7u << 22)        // pad_interval: 256 DWORDs
                | (3u << 25));      // pad_amount: 4 DWORDs
    g1[1] = (int)(256u << 16);      // tensor_dim0 = 256 (elements per row)
    g1[2] = (int)(16u << 16);       // tensor_dim0 hi = 0 ; tensor_dim1 = 16
    g1[3] = (int)(256u << 16);      // tensor_dim1 hi = 0 ; tile_dim0 = 256
    g1[4] = 16;                     // tile_dim1 = 16 ; tile_dim2 = 0
    g1[5] = 256;                    // tensor_dim0_stride = 256 (low 32)
    g1[6] = 0;                      // stride hi / dim1_stride
    g1[7] = 0;
    v4i z4 = {0, 0, 0, 0};
    v8i z8 = {0, 0, 0, 0, 0, 0, 0, 0};
    __builtin_amdgcn_tensor_load_to_lds(g0, g1, z4, z4, z8, 0);
}
#else
#define HAVE_TDM 0
#endif

// ---------------- Kernel 2: fp32 WMMA distances + argmin ----------------
// grid = 256 blocks, block = 128 threads (4 waves). Wave w owns rows
// [blockIdx*64 + w*16, +16). Emb tiles double-buffered in LDS via TDM DMA
// issued by wave 0 (TENSORcnt), compute overlaps the DMA of the next tile.
__global__ __launch_bounds__(128, 1) void vq_argmin(const float* __restrict__ z,
                                                    const float* __restrict__ emb,
                                                    const float* __restrict__ enorm,
                                                    float* __restrict__ outIdx) {
    __shared__ float sB[2][16 * LDSROW];

    const int lane  = threadIdx.x & 31;
    const int wave  = threadIdx.x >> 5;
    const int m     = lane & 15;               // A-matrix row within strip / B-matrix column
    const int khalf = (lane & 16) ? 2 : 0;     // K sub-offset per ISA 16x4 A layout
    const int strip = blockIdx.x * 64 + wave * 16;

    // ---- Preload A fragments: X[strip+m][*] in z's [B,C,H,W] layout ----
    // flat row n -> b = n>>10, hw = n&1023 ; element c at z + b*262144 + c*1024 + hw
    const int nrow = strip + m;
    const float* zrow = z + (size_t)(nrow >> 10) * 262144 + (size_t)(nrow & 1023);
    v2f afrag[64];
#pragma unroll
    for (int kk = 0; kk < 64; ++kk) {
        int c = 4 * kk + khalf;
        afrag[kk].x = zrow[(size_t)c * 1024];
        afrag[kk].y = zrow[(size_t)(c + 1) * 1024];
    }

    float best[8];
    int   bidx[8];
#pragma unroll
    for (int j = 0; j < 8; ++j) { best[j] = 3.4e38f; bidx[j] = 0; }

#if HAVE_TDM
    const unsigned ldsOff[2] = { (unsigned)(uintptr_t)&sB[0][0],
                                 (unsigned)(uintptr_t)&sB[1][0] };
    if (wave == 0) {
        tdm_load_tile(emb, ldsOff[0]);
        __builtin_amdgcn_s_wait_tensorcnt(0);
    }
    __syncthreads();
#endif

    for (int t = 0; t < NTILE; ++t) {
        const int s = t & 1;

#if HAVE_TDM
        // DMA next tile into the other buffer while we compute this one
        if (wave == 0 && t + 1 < NTILE)
            tdm_load_tile(emb + (size_t)(t + 1) * 4096, ldsOff[(t + 1) & 1]);
#else
        // Fallback: direct cooperative staging (two barriers per tile)
        {
            const float4* src = (const float4*)emb + (size_t)t * 1024;
#pragma unroll
            for (int i = 0; i < 8; ++i) {
                int cidx = threadIdx.x + i * 128;
                float4 v = src[cidx];
                int row = cidx >> 6, col4 = cidx & 63;
                *(float4*)&sB[s][row * LDSROW + col4 * 4] = v;
            }
        }
        __syncthreads();
#endif

        // ---- 16x16 distance tile via 64 x v_wmma_f32_16x16x4_f32 ----
        v8f acc = {0.f, 0.f, 0.f, 0.f, 0.f, 0.f, 0.f, 0.f};
        const float* Lb = &sB[s][m * LDSROW + khalf];  // B[k][n]: n=lane&15, k=khalf+{0,1}
#pragma unroll
        for (int kk = 0; kk < 64; ++kk) {
            v2f bf = *(const v2f*)(Lb + 4 * kk);
            acc = __builtin_amdgcn_wmma_f32_16x16x4_f32(
                false, afrag[kk], false, bf, (short)0, acc, false, false);
        }

        // dist = ||e||^2 - 2 x.e  (||x||^2 irrelevant for argmin)
        const int   ncode = t * 16 + m;       // code index for this lane (C-layout N = lane&15)
        const float en    = enorm[ncode];
#pragma unroll
        for (int j = 0; j < 8; ++j) {
            float d = en - 2.0f * acc[j];
            if (d < best[j]) { best[j] = d; bidx[j] = ncode; }
        }

#if HAVE_TDM
        if (wave == 0 && t + 1 < NTILE)
            __builtin_amdgcn_s_wait_tensorcnt(0);   // DMA done before barrier releases readers
#endif
        __syncthreads();
    }

    // ---- Butterfly argmin across the 16 lanes sharing each output row ----
    // C/D layout: VGPR j <-> row j (lanes 0-15) / row j+8 (lanes 16-31)
#pragma unroll
    for (int j = 0; j < 8; ++j) {
        float b = best[j]; int ix = bidx[j];
#pragma unroll
        for (int mm = 8; mm >= 1; mm >>= 1) {
            float ob = __shfl_xor(b, mm, 32);
            int   oi = __shfl_xor(ix, mm, 32);
            if (ob < b || (ob == b && oi < ix)) { b = ob; ix = oi; }
        }
        if (lane == 0)  outIdx[strip + j]     = (float)ix;
        if (lane == 16) outIdx[strip + 8 + j] = (float)ix;
    }
}

// ---------------- Kernel 3: gather codes -> output, per-block sq-err ----------------
__global__ void vq_gather(const float* __restrict__ emb, const float* __restrict__ z,
                          const float* __restrict__ idxF, float* __restrict__ out,
                          float* __restrict__ blocksum) {
    int e  = blockIdx.x * 256 + threadIdx.x;   // flat [B,C,H,W] element
    int b  = e >> 18;
    int c  = (e >> 10) & 255;
    int hw = e & 1023;
    int n  = (b << 10) | hw;
    int idx = (int)idxF[n];
    float q  = emb[(size_t)idx * CDIM + c];
    float zt = z[e];
    out[e] = q;                                 // straight-through value == quantized
    float d = q - zt;
    float s = d * d;
#pragma unroll
    for (int mm = 16; mm >= 1; mm >>= 1) s += __shfl_xor(s, mm, 32);
    __shared__ float red[8];
    int lane = threadIdx.x & 31, wave = threadIdx.x >> 5;
    if (lane == 0) red[wave] = s;
    __syncthreads();
    if (threadIdx.x == 0) {
        float tot = 0.f;
#pragma unroll
        for (int i = 0; i < 8; ++i) tot += red[i];
        blocksum[blockIdx.x] = tot;
    }
}

// ---------------- Kernel 4: deterministic final reduce -> losses ----------------
__global__ void vq_loss(const float* __restrict__ blocksum, float* __restrict__ out) {
    __shared__ float r[256];
    float s = 0.f;
    for (int k = 0; k < 64; ++k) s += blocksum[threadIdx.x + 256 * k];
    r[threadIdx.x] = s;
    __syncthreads();
    for (int off = 128; off > 0; off >>= 1) {
        if (threadIdx.x < off) r[threadIdx.x] += r[threadIdx.x + off];
        __syncthreads();
    }
    if (threadIdx.x == 0) {
        float mean = r[0] / (float)QELEMS;
        out[QELEMS]     = mean;          // codebook loss
        out[QELEMS + 1] = 0.25f * mean;  // commitment loss (BETA * same mean)
    }
}

extern "C" void kernel_launch(void* const* d_in, const int* in_sizes, int n_in,
                              void* d_out, int out_size, void* d_ws, size_t ws_size,
                              hipStream_t stream) {
    const float* z   = (const float*)d_in[0];
    const float* emb = (const float*)d_in[1];
    float* out    = (float*)d_out;
    float* ws     = (float*)d_ws;
    float* enorm  = ws;                  // 8192 floats
    float* bsum   = ws + KCODE;          // 16384 floats
    float* outIdx = out + IDX_OFF;       // 16384 floats (indices as float)

    vq_enorm <<<KCODE / 8, 256, 0, stream>>>(emb, enorm);
    vq_argmin<<<NPOS / 64, 128, 0, stream>>>(z, emb, enorm, outIdx);
    vq_gather<<<QELEMS / 256, 256, 0, stream>>>(emb, z, outIdx, out, bsum);
    vq_loss  <<<1, 256, 0, stream>>>(bsum, out);
}